// CachedMultiheadAttention_44573170598138
// MI455X (gfx1250) — compile-verified
//
#include <hip/hip_runtime.h>
#include <hip/hip_bf16.h>

// CachedMultiheadAttention for MI455X (gfx1250, wave32).
// Memory-bound (537MB fp32 K/V stream vs 2.7 GFLOP -> ~23us HBM floor at
// 23.3 TB/s). Single-pass flash-style attention in fp32 using
// V_WMMA_F32_16X16X4_F32, with K/V chunks staged into wave-private LDS by the
// Tensor Data Mover (tensor_load_to_lds, TENSORcnt) -- no block barriers in
// the streaming loop.

typedef float v2f __attribute__((ext_vector_type(2)));
typedef float v8f __attribute__((ext_vector_type(8)));
typedef unsigned int u32x4 __attribute__((ext_vector_type(4)));
typedef int i32x4 __attribute__((ext_vector_type(4)));
typedef int i32x8 __attribute__((ext_vector_type(8)));

#define T_LEN 8
#define BSZ 16
#define NHEAD 16
#define HDIM 64
#define SLEN 4096
#define HIDDEN 1024
#define SCALE_F 0.125f
#define MASK_INF 3.402823466e38f
#define WAVES 4

static __device__ __forceinline__ v8f wmma4(v2f a, v2f b, v8f c) {
  // D = A(16x4,f32) * B(4x16,f32) + C(16x16,f32)
  return __builtin_amdgcn_wmma_f32_16x16x4_f32(false, a, false, b, (short)0, c,
                                               false, false);
}

// TDM: DMA one contiguous 4KB tile (1024 f32 = one 16x64 K/V chunk) from
// global memory into LDS. D# layout per CDNA5 ISA ch.8. This toolchain uses
// the 6-arg builtin (g0, g1, g2, g3, g4, cpol).
static __device__ __forceinline__ void tdm_load_4kb(unsigned int lds_byte_addr,
                                                    const float* gptr,
                                                    i32x8 g1, i32x4 z4,
                                                    i32x8 z8) {
  unsigned long long ga = (unsigned long long)(uintptr_t)gptr;
  u32x4 g0;
  g0[0] = 1u;                    // count=1 (valid user descriptor)
  g0[1] = lds_byte_addr;         // lds_addr (bytes)
  g0[2] = (unsigned int)ga;      // global_addr[31:0]
  g0[3] = ((unsigned int)(ga >> 32) & 0x01FFFFFFu) | (2u << 30);  // [56:32]|type=2
  __builtin_amdgcn_tensor_load_to_lds(g0, g1, z4, z4, z8, 0);
}

// ---------------------------------------------------------------------------
// Kernel 1: q = (X @ Wq.T + bq) * scale, written to q_ws[b][h][t][hd]
// X rows r = t*16+b (128 rows), cols 1024. One wave per 16x16 tile.
__global__ void cmha_qproj(const float* __restrict__ X,
                           const float* __restrict__ Wq,
                           const float* __restrict__ bq,
                           float* __restrict__ q_ws) {
  const int lane = threadIdx.x & 31;
  const int tile_m = blockIdx.x >> 6;  // 0..7
  const int tile_n = blockIdx.x & 63;  // 0..63
  const int ml = lane & 15;
  const int c = (lane >> 4) << 1;  // lanes 0-15 -> K+{0,1}; lanes 16-31 -> K+{2,3}
  const float* arow = X + (size_t)(tile_m * 16 + ml) * HIDDEN + c;
  const float* brow = Wq + (size_t)(tile_n * 16 + ml) * HIDDEN + c;
  v8f acc = {};
#pragma unroll 8
  for (int k = 0; k < HIDDEN / 4; ++k) {
    v2f a = *(const v2f*)(arow + 4 * k);
    v2f b = *(const v2f*)(brow + 4 * k);
    acc = wmma4(a, b, acc);
  }
  const float bias = bq[tile_n * 16 + ml];
  const int mbase = tile_m * 16 + ((lane >> 4) << 3);
  const int col = tile_n * 16 + ml;
  const int h = col >> 6, hd = col & 63;
#pragma unroll
  for (int j = 0; j < 8; ++j) {
    const int r = mbase + j;  // r = t*16 + b
    const int t = r >> 4, b = r & 15;
    q_ws[(((size_t)(b * NHEAD + h) * T_LEN) + t) * HDIM + hd] =
        (acc[j] + bias) * SCALE_F;
  }
}

// ---------------------------------------------------------------------------
// Kernel 2: fused gather + QK^T + mask + online softmax + PV.
// One block (4 waves) per (b,h); each wave streams interleaved 16-wide S
// chunks with TDM staging into its private LDS region (no block barriers).
__global__ __launch_bounds__(WAVES * 32) void cmha_attn(
    const float* __restrict__ q_ws, const float* __restrict__ K,
    const float* __restrict__ V, const int* __restrict__ mask,
    const int* __restrict__ order, float* __restrict__ attn_ws) {
  __shared__ __align__(16) float k_lds[WAVES][16 * HDIM];
  __shared__ __align__(16) float v_lds[WAVES][16 * HDIM];
  __shared__ __align__(16) float p_lds[WAVES][16 * 16];
  __shared__ __align__(16) float acc_lds[WAVES][8 * HDIM];
  __shared__ float s_m[WAVES][8];
  __shared__ float s_l[WAVES][8];

  const int bh = blockIdx.x;  // b*16 + h
  const int b = bh >> 4, h = bh & 15;
  const int wave = threadIdx.x >> 5;
  const int lane = threadIdx.x & 31;
  const int ml = lane & 15;
  const int c = (lane >> 4) << 1;

  const int srcb = order[b];
  const float* Kb = K + (size_t)(srcb * NHEAD + h) * SLEN * HDIM;
  const float* Vb = V + (size_t)(srcb * NHEAD + h) * SLEN * HDIM;
  const int* mrow = mask + (size_t)b * SLEN;

  // TDM descriptor group1 (shared by K and V tiles): 1024 x f32, 1 row.
  i32x8 g1;
  g1[0] = (int)(2u << 16);     // data_size = 4 bytes
  g1[1] = (int)(1024u << 16);  // tensor_dim0 = 1024 (bits[79:48] low half)
  g1[2] = (int)(1u << 16);     // tensor_dim1 = 1    (bits[111:80] low half)
  g1[3] = (int)(1024u << 16);  // tile_dim0 = 1024   (bits[127:112])
  g1[4] = 1;                   // tile_dim1 = 1, tile_dim2 = 0
  g1[5] = 1024;                // tensor_dim0_stride = 1024
  g1[6] = 0;
  g1[7] = 0;
  i32x4 z4 = {0, 0, 0, 0};
  i32x8 z8 = {0, 0, 0, 0, 0, 0, 0, 0};

  float* kw = k_lds[wave];
  float* vw = v_lds[wave];
  float* pw = p_lds[wave];
  const unsigned int kw_lds = (unsigned int)(uintptr_t)kw;
  const unsigned int vw_lds = (unsigned int)(uintptr_t)vw;

  // Q as WMMA A-operand: lane holds row m=lane&15, K-pairs {4i+c, 4i+c+1}.
  // Rows 8..15 padded to zero (T=8).
  v2f qa[16];
#pragma unroll
  for (int i = 0; i < 16; ++i) {
    if (ml < T_LEN) {
      qa[i] = *(const v2f*)(q_ws + ((size_t)bh * T_LEN + ml) * HDIM + 4 * i + c);
    } else {
      v2f z; z[0] = 0.0f; z[1] = 0.0f;
      qa[i] = z;
    }
  }

  v8f acc0 = {}, acc1 = {}, acc2 = {}, acc3 = {};
  float rm[8], rl[8];
#pragma unroll
  for (int j = 0; j < 8; ++j) { rm[j] = -3.0e38f; rl[j] = 0.0f; }

  for (int chunk = wave; chunk < SLEN / 16; chunk += WAVES) {
    const int s0 = chunk * 16;
    if (chunk + WAVES < SLEN / 16) {  // hint next chunk toward L2
      __builtin_prefetch(Kb + (size_t)(s0 + WAVES * 16) * HDIM + lane * 16, 0, 3);
      __builtin_prefetch(Vb + (size_t)(s0 + WAVES * 16) * HDIM + lane * 16, 0, 3);
    }
    // All prior LDS reads of kw/vw were consumed last iteration; make the WAR
    // explicit before the DMA engine overwrites the buffers.
    asm volatile("s_wait_dscnt 0x0" ::: "memory");
    tdm_load_4kb(kw_lds, Kb + (size_t)s0 * HDIM, g1, z4, z8);
    tdm_load_4kb(vw_lds, Vb + (size_t)s0 * HDIM, g1, z4, z8);
    __builtin_amdgcn_s_wait_tensorcnt(0);
    __builtin_amdgcn_wave_barrier();

    // scores tile (16x16): D = Q(16x64) * K^T(64x16). 16 chained K=4 WMMAs.
    v8f sc = {};
#pragma unroll
    for (int i = 0; i < 16; ++i) {
      v2f kb = *(const v2f*)(kw + ml * HDIM + 4 * i + c);
      sc = wmma4(qa[i], kb, sc);
    }

    // Mask: column s = s0 + ml (N = lane&15 in D layout).
    const float msub = MASK_INF * (float)mrow[s0 + ml];
    // Online softmax per row j (rows 0..7 live in lanes 0-15).
    float pvals[8];
#pragma unroll
    for (int j = 0; j < 8; ++j) {
      float v = sc[j] - msub;
      float mx = v;
      mx = fmaxf(mx, __shfl_xor(mx, 1, 32));
      mx = fmaxf(mx, __shfl_xor(mx, 2, 32));
      mx = fmaxf(mx, __shfl_xor(mx, 4, 32));
      mx = fmaxf(mx, __shfl_xor(mx, 8, 32));
      const float newm = fmaxf(rm[j], mx);
      const float scale = __expf(rm[j] - newm);
      const float p = __expf(v - newm);
      float ps = p;
      ps += __shfl_xor(ps, 1, 32);
      ps += __shfl_xor(ps, 2, 32);
      ps += __shfl_xor(ps, 4, 32);
      ps += __shfl_xor(ps, 8, 32);
      rl[j] = rl[j] * scale + ps;
      rm[j] = newm;
      pvals[j] = p;
      acc0[j] *= scale; acc1[j] *= scale; acc2[j] *= scale; acc3[j] *= scale;
    }

    // Transpose p through wave-private LDS into A-operand layout.
    // (LDS ops from one wave execute in order; wave_barrier pins scheduling.)
#pragma unroll
    for (int j = 0; j < 8; ++j) {
      pw[(j + ((lane >> 4) << 3)) * 16 + ml] = pvals[j];
    }
    __builtin_amdgcn_wave_barrier();

    // PV: out(16x64) += P(16x16) * V(16x64); 4 K-steps x 4 hd tiles.
    v2f pa[4];
#pragma unroll
    for (int i = 0; i < 4; ++i) {
      pa[i] = *(const v2f*)(pw + ml * 16 + 4 * i + c);
    }
#pragma unroll
    for (int i = 0; i < 4; ++i) {
      const float* v0 = vw + (4 * i + c) * HDIM + ml;
      const float* v1 = v0 + HDIM;
      v2f b0; b0[0] = v0[0];  b0[1] = v1[0];
      v2f b1; b1[0] = v0[16]; b1[1] = v1[16];
      v2f b2; b2[0] = v0[32]; b2[1] = v1[32];
      v2f b3; b3[0] = v0[48]; b3[1] = v1[48];
      acc0 = wmma4(pa[i], b0, acc0);
      acc1 = wmma4(pa[i], b1, acc1);
      acc2 = wmma4(pa[i], b2, acc2);
      acc3 = wmma4(pa[i], b3, acc3);
    }
  }

  // ---- merge the 4 per-wave partial softmaxes ----
  __syncthreads();
  if (lane == 0) {
#pragma unroll
    for (int j = 0; j < 8; ++j) { s_m[wave][j] = rm[j]; s_l[wave][j] = rl[j]; }
  }
  __syncthreads();
  if (lane < 16) {
#pragma unroll
    for (int j = 0; j < 8; ++j) {
      float gm = s_m[0][j];
#pragma unroll
      for (int w = 1; w < WAVES; ++w) gm = fmaxf(gm, s_m[w][j]);
      const float f = __expf(rm[j] - gm);
      acc_lds[wave][j * HDIM + 0 + lane]  = acc0[j] * f;
      acc_lds[wave][j * HDIM + 16 + lane] = acc1[j] * f;
      acc_lds[wave][j * HDIM + 32 + lane] = acc2[j] * f;
      acc_lds[wave][j * HDIM + 48 + lane] = acc3[j] * f;
    }
  }
  __syncthreads();
#pragma unroll
  for (int e = 0; e < 4; ++e) {
    const int idx = threadIdx.x + e * (WAVES * 32);  // 0..511
    const int m = idx >> 6, hd = idx & 63;
    float gm = s_m[0][m];
#pragma unroll
    for (int w = 1; w < WAVES; ++w) gm = fmaxf(gm, s_m[w][m]);
    float gl = 0.0f, sum = 0.0f;
#pragma unroll
    for (int w = 0; w < WAVES; ++w) {
      gl += s_l[w][m] * __expf(s_m[w][m] - gm);
      sum += acc_lds[w][m * HDIM + hd];
    }
    attn_ws[((size_t)bh * T_LEN + m) * HDIM + hd] = sum / gl;
  }
}

// ---------------------------------------------------------------------------
// Kernel 3: out = attn_out @ Wo.T + bo, reading attn_ws[b][h][t][hd].
__global__ void cmha_oproj(const float* __restrict__ attn_ws,
                           const float* __restrict__ Wo,
                           const float* __restrict__ bo,
                           float* __restrict__ out) {
  const int lane = threadIdx.x & 31;
  const int tile_m = blockIdx.x >> 6;
  const int tile_n = blockIdx.x & 63;
  const int ml = lane & 15;
  const int c = (lane >> 4) << 1;
  const int r = tile_m * 16 + ml;  // r = t*16 + b
  const int t = r >> 4, b = r & 15;
  const float* brow = Wo + (size_t)(tile_n * 16 + ml) * HIDDEN + c;
  v8f acc = {};
#pragma unroll 4
  for (int k = 0; k < HIDDEN / 4; ++k) {
    const int kc = 4 * k + c;  // even -> never crosses a head boundary
    const int h = kc >> 6, hd = kc & 63;
    v2f a = *(const v2f*)(attn_ws + ((size_t)(b * NHEAD + h) * T_LEN + t) * HDIM + hd);
    v2f bb = *(const v2f*)(brow + 4 * k);
    acc = wmma4(a, bb, acc);
  }
  const float bias = bo[tile_n * 16 + ml];
  const int mbase = tile_m * 16 + ((lane >> 4) << 3);
  const int col = tile_n * 16 + ml;
#pragma unroll
  for (int j = 0; j < 8; ++j) {
    out[(size_t)(mbase + j) * HIDDEN + col] = acc[j] + bias;
  }
}

// ---------------------------------------------------------------------------
extern "C" void kernel_launch(void* const* d_in, const int* in_sizes, int n_in,
                              void* d_out, int out_size, void* d_ws,
                              size_t ws_size, hipStream_t stream) {
  const float* X     = (const float*)d_in[0];  // (T,B,HIDDEN)
  const float* K     = (const float*)d_in[1];  // (B,H,S,HD)
  const float* V     = (const float*)d_in[2];  // (B,H,S,HD)
  const int*   mask  = (const int*)d_in[3];    // (B,S)
  const int*   order = (const int*)d_in[4];    // (B,)
  const float* Wq    = (const float*)d_in[5];
  const float* bq    = (const float*)d_in[6];
  const float* Wo    = (const float*)d_in[7];
  const float* bo    = (const float*)d_in[8];
  float* out = (float*)d_out;

  float* q_ws = (float*)d_ws;                                  // 128K floats
  float* attn_ws = q_ws + (size_t)BSZ * NHEAD * T_LEN * HDIM;  // 128K floats

  cmha_qproj<<<512, 32, 0, stream>>>(X, Wq, bq, q_ws);
  cmha_attn<<<BSZ * NHEAD, WAVES * 32, 0, stream>>>(q_ws, K, V, mask, order,
                                                    attn_ws);
  cmha_oproj<<<512, 32, 0, stream>>>(attn_ws, Wo, bo, out);
}